// CenterWoParamsLoss_15917148799632
// MI455X (gfx1250) — compile-verified
//
#include <hip/hip_runtime.h>

#define BATCH 16384
#define FEAT  2048

typedef float v2f __attribute__((ext_vector_type(2)));
typedef float v4f __attribute__((ext_vector_type(4)));
typedef float v8f __attribute__((ext_vector_type(8)));

// Reduce 32 per-lane f32 partial sums to a single wave total using
// V_WMMA_F32_16X16X4_F32 with B == all-ones.
// A layout (16x4 f32): VGPR0 = A[m,0] (lanes 0-15) / A[m,2] (lanes 16-31),
// VGPR1 = A[m,1]/A[m,3] -> set to 0.  With B[k,n] == 1:
//   D[m,n] = acc[lane m] + acc[lane m+16]  (replicated across n)
// D layout: VGPR i holds M=i (lanes 0-15) and M=i+8 (lanes 16-31), so
// sum(d[0..7]) per lane + shfl_xor(16) gives the full 32-lane sum in all lanes.
__device__ __forceinline__ float wave_reduce_wmma(float acc) {
  v2f a; a[0] = acc;  a[1] = 0.0f;
  v2f b; b[0] = 1.0f; b[1] = 1.0f;
  v8f c = {};
  v8f d = __builtin_amdgcn_wmma_f32_16x16x4_f32(
      /*neg_a=*/false, a, /*neg_b=*/false, b,
      /*c_mod=*/(short)0, c, /*reuse_a=*/false, /*reuse_b=*/false);
  float s = d[0] + d[1] + d[2] + d[3] + d[4] + d[5] + d[6] + d[7];
  s += __shfl_xor(s, 16, 32);   // combine M=0..7 half with M=8..15 half
  return s;                     // wave total, present in every lane
}

__global__ __launch_bounds__(256) void center_loss_partial(
    const float* __restrict__ x, const long long* __restrict__ labels,
    const float* __restrict__ centers, float* __restrict__ partials) {
  const int t = threadIdx.x;
  float acc = 0.0f;

  for (int r = blockIdx.x; r < BATCH; r += gridDim.x) {
    // Prefetch this block's next x row (streamed once; covers latency).
    int nr = r + gridDim.x;
    if (nr < BATCH) {
      const char* px = (const char*)(x + (size_t)nr * FEAT);
      __builtin_prefetch(px + (size_t)t * 32, 0, 0);  // global_prefetch_b8
    }

    const int lbl = (int)labels[r];   // same addr across lanes -> broadcast
    const v4f* xr = (const v4f*)(x + (size_t)r * FEAT);
    const v4f* cr = (const v4f*)(centers + (size_t)lbl * FEAT);

    // 256 threads * 2 float4 = 2048 floats = one full row, coalesced b128.
    // x is read-once: non-temporal so it doesn't evict centers from L2.
    v4f a0 = __builtin_nontemporal_load(xr + t);
    v4f a1 = __builtin_nontemporal_load(xr + t + 256);
    v4f c0 = cr[t];
    v4f c1 = cr[t + 256];

    v4f d0 = a0 - c0;
    v4f d1 = a1 - c1;
    acc += d0[0]*d0[0] + d0[1]*d0[1] + d0[2]*d0[2] + d0[3]*d0[3]
         + d1[0]*d1[0] + d1[1]*d1[1] + d1[2]*d1[2] + d1[3]*d1[3];
  }

  // Block reduction: 8 waves -> WMMA wave reduce -> LDS -> WMMA wave reduce.
  __shared__ float red[8];
  const int wave = t >> 5, lane = t & 31;
  float w = wave_reduce_wmma(acc);
  if (lane == 0) red[wave] = w;
  __syncthreads();
  if (wave == 0) {                       // EXEC all-ones within wave 0
    float v = (lane < 8) ? red[lane] : 0.0f;
    v = wave_reduce_wmma(v);
    if (lane == 0) partials[blockIdx.x] = v;
  }
}

__global__ __launch_bounds__(1024) void center_loss_final(
    const float* __restrict__ partials, int n, float* __restrict__ out) {
  const int t = threadIdx.x;
  float acc = 0.0f;
  for (int i = t; i < n; i += 1024) acc += partials[i];

  __shared__ float red[32];
  const int wave = t >> 5, lane = t & 31;
  float w = wave_reduce_wmma(acc);
  if (lane == 0) red[wave] = w;
  __syncthreads();
  if (wave == 0) {
    float v = red[lane];                 // exactly 32 wave partials
    v = wave_reduce_wmma(v);
    if (lane == 0) out[0] = v * (0.5f / (float)BATCH);
  }
}

extern "C" void kernel_launch(void* const* d_in, const int* in_sizes, int n_in,
                              void* d_out, int out_size, void* d_ws, size_t ws_size,
                              hipStream_t stream) {
  const float*     x       = (const float*)d_in[0];
  const long long* labels  = (const long long*)d_in[1];   // int64 labels
  const float*     centers = (const float*)d_in[2];
  float*           outp    = (float*)d_out;
  float*           partials = (float*)d_ws;

  int nblocks = 4096;                                     // 4 rows per block
  if (ws_size < (size_t)nblocks * sizeof(float))
    nblocks = (int)(ws_size / sizeof(float));
  if (nblocks < 1) nblocks = 1;

  center_loss_partial<<<nblocks, 256, 0, stream>>>(x, labels, centers, partials);
  center_loss_final<<<1, 1024, 0, stream>>>(partials, nblocks, outp);
}